// GFlowNetActor_2448131359392
// MI455X (gfx1250) — compile-verified
//
#include <hip/hip_runtime.h>
#include <hip/hip_bf16.h>
#include <math.h>

typedef __attribute__((ext_vector_type(2))) float v2f;
typedef __attribute__((ext_vector_type(8))) float v8f;

#define B_TOT   4096
#define NTOTAL  100000
#define DDEG    32
#define HDIM    64
#define TSTEPS  8
#define ROWS    16            // rollouts per workgroup
#define NEGV    (-1e9f)
#define LN_EPS  1e-5f

// D = A(16x64) x W(64x64) tile GEMM on V_WMMA_F32_16X16X4_F32.
// Fragment layouts per CDNA5 ISA 7.12.2 (32-bit A 16x4 / B 4x16 / C-D 16x16).

__launch_bounds__(128)
__global__ void gflownet_rollout_kernel(
    const float* __restrict__ q_tokens,     // [B,H]
    const int*   __restrict__ nbr_node,     // [NTOT,D]
    const int*   __restrict__ start_nodes,  // [B]
    const int*   __restrict__ edge_valid,   // [NTOT,D] (0/1)
    const float* __restrict__ action_keys,  // [NTOT,D,H]
    const float* __restrict__ W_init,       // [H,H]
    const float* __restrict__ W_h,          // [H,H]
    const float* __restrict__ W_k,          // [H,H]
    const float* __restrict__ ln_gamma,     // [H+2]
    const float* __restrict__ ln_beta,      // [H+2]
    const float* __restrict__ stop_w,       // [H+2]
    const float* __restrict__ stop_b,       // scalar
    const float* __restrict__ temperature,  // scalar
    int*   __restrict__ actions_out,        // [B,T+1] int32
    float* __restrict__ lp_total_out,       // [B]
    float* __restrict__ lp_steps_out,       // [B,T+1]
    int*   __restrict__ nmoves_out)         // [B] int32
{
    __shared__ float sHidden[ROWS * HDIM];  // agent hidden state tile
    __shared__ float sIn[ROWS * HDIM];      // q tile, later selected-key tile
    __shared__ float sW0[HDIM * HDIM];      // W_init, later W_h
    __shared__ float sW1[HDIM * HDIM];      // W_k
    __shared__ float sScores[ROWS * DDEG];
    __shared__ int   sMask[ROWS];
    __shared__ int   sCurr[ROWS], sCurrNext[ROWS];
    __shared__ int   sStopped[ROWS], sStoppedNext[ROWS];
    __shared__ int   sKeep[ROWS];
    __shared__ int   sAct[ROWS];

    const int tid   = threadIdx.x;
    const int wave  = tid >> 5;      // 0..3 : N-tile owner
    const int lane  = tid & 31;
    const int m     = lane & 15;     // M (A/D) or N (B/D) index within tile
    const int khi   = lane >> 4;     // K-half selector for A/B fragments
    const int base  = blockIdx.x * ROWS;

    // ---- stage W_init + q tile (B128 path); init per-rollout state ----
    {
        const float4* w4 = (const float4*)W_init;
        float4*       s4 = (float4*)sW0;
        for (int i = tid; i < (HDIM * HDIM) / 4; i += 128) s4[i] = w4[i];
        const float4* q4 = (const float4*)(q_tokens + (size_t)base * HDIM);
        float4*       i4 = (float4*)sIn;
        for (int i = tid; i < (ROWS * HDIM) / 4; i += 128) i4[i] = q4[i];
    }
    if (tid < ROWS) {
        sCurr[tid]    = start_nodes[base + tid];
        sStopped[tid] = 0;
    }
    __syncthreads();

    // ---- hidden0 = tanh(q @ W_init) : WMMA f32 16x16x4 chain ----
    {
        v8f acc = {0.f, 0.f, 0.f, 0.f, 0.f, 0.f, 0.f, 0.f};
        for (int k0 = 0; k0 < HDIM; k0 += 4) {
            const int k = k0 + 2 * khi;
            v2f a, b;
            a.x = sIn[m * HDIM + k];
            a.y = sIn[m * HDIM + k + 1];
            b.x = sW0[k * HDIM + wave * 16 + m];
            b.y = sW0[(k + 1) * HDIM + wave * 16 + m];
            acc = __builtin_amdgcn_wmma_f32_16x16x4_f32(
                false, a, false, b, (short)0, acc, false, false);
        }
        __syncthreads();   // all A/B reads done before D writeback
        for (int r = 0; r < 8; ++r) {
            int row = r + 8 * khi;
            sHidden[row * HDIM + wave * 16 + m] = tanhf(acc[r]);
        }
    }
    // stage W_h / W_k (sW0 reads finished above), B128 path
    {
        const float4* wh4 = (const float4*)W_h;
        const float4* wk4 = (const float4*)W_k;
        float4* s04 = (float4*)sW0;
        float4* s14 = (float4*)sW1;
        for (int i = tid; i < (HDIM * HDIM) / 4; i += 128) {
            s04[i] = wh4[i];
            s14[i] = wk4[i];
        }
    }

    float lp_total = 0.f;
    int   nmoves   = 0;
    const float tclamp = fmaxf(temperature[0], 1e-5f);
    const float sbias  = stop_b[0];

    for (int t = 0; t <= TSTEPS; ++t) {
        if (tid < ROWS) sMask[tid] = 0;
        __syncthreads();   // (A) weights/hidden/state visible

        // ---- scores: 512 (rollout,edge) pairs; key rows via B128 loads ----
        for (int idx = tid; idx < ROWS * DDEG; idx += 128) {
            int r = idx >> 5, d = idx & 31;
            int node = sCurr[r];
            bool valid = (!sStopped[r]) && (t < TSTEPS) &&
                         (edge_valid[(size_t)node * DDEG + d] != 0);
            float sc = NEGV;
            if (valid) {
                const float4* kp4 =
                    (const float4*)(action_keys + ((size_t)node * DDEG + d) * HDIM);
                const float4* h4 = (const float4*)(sHidden + r * HDIM);
                float dot = 0.f;
                #pragma unroll
                for (int h = 0; h < HDIM / 4; ++h) {
                    float4 kv = kp4[h];
                    float4 hv = h4[h];
                    dot += hv.x * kv.x + hv.y * kv.y + hv.z * kv.z + hv.w * kv.w;
                }
                sc = dot;
                atomicOr(&sMask[r], 1);
            }
            sScores[r * DDEG + d] = sc;
        }
        __syncthreads();   // (B)

        // ---- decision (LN + stop logit + log-softmax + argmax), 16 lanes ----
        if (tid < ROWS) {
            const int r = tid, g = base + r;
            const int stopped = sStopped[r];
            const int curr = sCurr[r];
            int action; float lp;
            if (stopped) { action = DDEG; lp = 0.f; }
            else {
                const float step_frac = (float)t / (float)TSTEPS;
                const float has_edge  = sMask[r] ? 1.f : 0.f;
                float mu = step_frac + has_edge;
                for (int i = 0; i < HDIM; ++i) mu += sHidden[r * HDIM + i];
                mu *= (1.f / (float)(HDIM + 2));
                float var = 0.f;
                for (int i = 0; i < HDIM; ++i) {
                    float xc = sHidden[r * HDIM + i] - mu; var += xc * xc;
                }
                { float xc = step_frac - mu; var += xc * xc; }
                { float xc = has_edge  - mu; var += xc * xc; }
                var *= (1.f / (float)(HDIM + 2));
                const float inv = rsqrtf(var + LN_EPS);
                float slogit = sbias;
                for (int i = 0; i < HDIM + 2; ++i) {
                    float xi = (i < HDIM) ? sHidden[r * HDIM + i]
                              : (i == HDIM ? step_frac : has_edge);
                    slogit += ((xi - mu) * inv * ln_gamma[i] + ln_beta[i]) * stop_w[i];
                }
                float logits[DDEG + 1];
                for (int d = 0; d < DDEG; ++d) logits[d] = sScores[r * DDEG + d] / tclamp;
                logits[DDEG] = slogit / tclamp;
                float best = logits[0]; int bi = 0;
                for (int i = 1; i <= DDEG; ++i)
                    if (logits[i] > best) { best = logits[i]; bi = i; }
                float se = 0.f;
                for (int i = 0; i <= DDEG; ++i) se += expf(logits[i] - best);
                action = bi;
                lp = logits[bi] - (best + logf(se));
            }
            const bool chose_stop = (action == DDEG);
            const int a = (action < DDEG) ? action : (DDEG - 1);
            actions_out[(size_t)g * (TSTEPS + 1) + t] =
                chose_stop ? -1 : (curr * DDEG + a);
            lp_steps_out[(size_t)g * (TSTEPS + 1) + t] = lp;
            lp_total += lp;
            if (!chose_stop) nmoves++;
            sAct[r]  = a;
            sKeep[r] = chose_stop ? 1 : 0;
            const int tail = nbr_node[(size_t)curr * DDEG + a];
            sCurrNext[r]    = chose_stop ? curr : tail;
            sStoppedNext[r] = stopped | (chose_stop ? 1 : 0);
        }
        __syncthreads();   // (C)

        // ---- gather selected key rows (B128), zero for stopped rollouts ----
        for (int i = tid; i < (ROWS * HDIM) / 4; i += 128) {
            int r = i >> 4, h = i & 15;   // h indexes float4 within the row
            float4 v;
            if (sKeep[r]) {
                v.x = 0.f; v.y = 0.f; v.z = 0.f; v.w = 0.f;
            } else {
                const float4* kp4 =
                    (const float4*)(action_keys +
                                    ((size_t)sCurr[r] * DDEG + sAct[r]) * HDIM);
                v = kp4[h];
            }
            ((float4*)sIn)[i] = v;
        }
        __syncthreads();   // (D)

        // ---- new_hidden = tanh(hidden@W_h + sel@W_k) : fused WMMA chain ----
        {
            v8f acc = {0.f, 0.f, 0.f, 0.f, 0.f, 0.f, 0.f, 0.f};
            for (int k0 = 0; k0 < HDIM; k0 += 4) {
                const int k = k0 + 2 * khi;
                v2f a1, b1, a2, b2;
                a1.x = sHidden[m * HDIM + k];
                a1.y = sHidden[m * HDIM + k + 1];
                b1.x = sW0[k * HDIM + wave * 16 + m];
                b1.y = sW0[(k + 1) * HDIM + wave * 16 + m];
                acc = __builtin_amdgcn_wmma_f32_16x16x4_f32(
                    false, a1, false, b1, (short)0, acc, false, false);
                a2.x = sIn[m * HDIM + k];
                a2.y = sIn[m * HDIM + k + 1];
                b2.x = sW1[k * HDIM + wave * 16 + m];
                b2.y = sW1[(k + 1) * HDIM + wave * 16 + m];
                acc = __builtin_amdgcn_wmma_f32_16x16x4_f32(
                    false, a2, false, b2, (short)0, acc, false, false);
            }
            __syncthreads();   // (E) all sHidden reads done before writeback
            for (int r = 0; r < 8; ++r) {
                int row = r + 8 * khi;
                if (!sKeep[row])
                    sHidden[row * HDIM + wave * 16 + m] = tanhf(acc[r]);
            }
        }
        if (tid < ROWS) {
            sCurr[tid]    = sCurrNext[tid];
            sStopped[tid] = sStoppedNext[tid];
        }
        // loop-top barrier (A) orders these for the next step
    }

    if (tid < ROWS) {
        lp_total_out[base + tid] = lp_total;
        nmoves_out[base + tid]   = nmoves;
    }
}

extern "C" void kernel_launch(void* const* d_in, const int* in_sizes, int n_in,
                              void* d_out, int out_size, void* d_ws, size_t ws_size,
                              hipStream_t stream) {
    const float* q_tokens    = (const float*)d_in[0];
    const int*   nbr_node    = (const int*)  d_in[1];
    const int*   start_nodes = (const int*)  d_in[2];
    const int*   edge_valid  = (const int*)  d_in[3];
    const float* action_keys = (const float*)d_in[4];
    const float* W_init      = (const float*)d_in[5];
    const float* W_h         = (const float*)d_in[6];
    const float* W_k         = (const float*)d_in[7];
    const float* ln_gamma    = (const float*)d_in[8];
    const float* ln_beta     = (const float*)d_in[9];
    const float* stop_w      = (const float*)d_in[10];
    const float* stop_b      = (const float*)d_in[11];
    const float* temperature = (const float*)d_in[12];

    // Outputs concatenated flat in return order:
    // actions_seq[B,T+1] (i32), log_pf_total[B] (f32),
    // log_pf_steps[B,T+1] (f32), num_moves[B] (i32)
    float* outf = (float*)d_out;
    int*   actions_out  = (int*)outf;
    float* lp_total_out = outf + (size_t)B_TOT * (TSTEPS + 1);
    float* lp_steps_out = lp_total_out + B_TOT;
    int*   nmoves_out   = (int*)(lp_steps_out + (size_t)B_TOT * (TSTEPS + 1));

    dim3 grid(B_TOT / ROWS);   // 256 blocks
    dim3 block(128);           // 4 wave32
    gflownet_rollout_kernel<<<grid, block, 0, stream>>>(
        q_tokens, nbr_node, start_nodes, edge_valid, action_keys,
        W_init, W_h, W_k, ln_gamma, ln_beta, stop_w, stop_b, temperature,
        actions_out, lp_total_out, lp_steps_out, nmoves_out);
}